// densityModule_18468359373477
// MI455X (gfx1250) — compile-verified
//
#include <hip/hip_runtime.h>
#include <cstdint>

// ---------------------------------------------------------------------------
// SPH density: rho[dst[e]] += W(r[e], h) * areas[src[e]]
// CDNA5 strategy: wave-private async global->LDS double-buffered pipeline
// (ASYNCcnt / s_wait_asynccnt), LDS b128 consumption, hw f32 global atomics.
// ---------------------------------------------------------------------------

#define TILE 128          // edges per wave-tile (32 lanes x 4 edges)
#define WPB  8            // waves per 256-thread block (wave32)
#define BLOCK 256
#define MAX_BLOCKS 1024

#define AS1 __attribute__((address_space(1)))
#define AS3 __attribute__((address_space(3)))

#ifndef __has_builtin
#define __has_builtin(x) 0
#endif

typedef int v4i __attribute__((vector_size(4 * sizeof(int))));

// 16-byte-per-lane async copy: global -> LDS (tracked by ASYNCcnt)
__device__ __forceinline__ void async_copy16(void* lds, const void* g) {
#if __has_builtin(__builtin_amdgcn_global_load_async_to_lds_b128)
    __builtin_amdgcn_global_load_async_to_lds_b128(
        (AS1 v4i*)g, (AS3 v4i*)lds, /*offset=*/0, /*cpol=*/0);
#else
    asm volatile("global_load_async_to_lds_b128 %0, %1, off"
                 :: "v"((unsigned)(uintptr_t)(AS3 char*)lds),
                    "v"((unsigned long long)(uintptr_t)g)
                 : "memory");
#endif
}

#if __has_builtin(__builtin_amdgcn_s_wait_asynccnt)
#define WAIT_ASYNC(n) do { __builtin_amdgcn_s_wait_asynccnt(n); \
                           asm volatile("" ::: "memory"); } while (0)
#else
#define WAIT_ASYNC(n) asm volatile("s_wait_asynccnt %0" :: "i"(n) : "memory")
#endif

// Wendland C2 contribution for one edge, scattered with a non-returning
// hardware f32 atomic (L2-resident rho -> global_atomic_add_f32).
__device__ __forceinline__ void sph_edge(float q, int s, int d, float C,
                                         const float* __restrict__ areas,
                                         float* __restrict__ rho) {
    float b  = __builtin_fminf(__builtin_fmaxf(1.0f - q, 0.0f), 1.0f);
    float b2 = b * b;
    float w  = C * (b2 * b2) * __builtin_fmaf(4.0f, q, 1.0f);
    unsafeAtomicAdd(&rho[d], w * areas[s]);
}

__global__ void __launch_bounds__(BLOCK)
density_zero_kernel(float* __restrict__ out, int n) {
    int i = blockIdx.x * BLOCK + threadIdx.x;
    if (i < n) out[i] = 0.0f;
}

__global__ void __launch_bounds__(BLOCK)
density_main_kernel(const float* __restrict__ rdist,
                    const float* __restrict__ areas,
                    const int*   __restrict__ src,
                    const int*   __restrict__ dst,
                    const float* __restrict__ supportPtr,
                    float* __restrict__ rho,
                    int nTiles) {
    __shared__ __align__(16) float rB[WPB][2][TILE];
    __shared__ __align__(16) int   sB[WPB][2][TILE];
    __shared__ __align__(16) int   dB[WPB][2][TILE];

    const int lane = threadIdx.x & 31;          // wave32
    const int wv   = threadIdx.x >> 5;
    const int wave   = blockIdx.x * WPB + wv;   // uniform per wave
    const int nWaves = gridDim.x * WPB;

    const float h = *supportPtr;
    const float C = (7.0f / 3.14159265358979323846f) / (h * h);

    int t = wave;
    if (t >= nTiles) return;                    // whole wave exits together

    const int lo = lane * 4;                    // 4 edges / lane, 16B aligned

    // ---- prologue: stage tile t into buffer 0 -----------------------------
    {
        long long e0 = (long long)t * TILE + lo;
        async_copy16(&rB[wv][0][lo], rdist + e0);
        async_copy16(&sB[wv][0][lo], src   + e0);
        async_copy16(&dB[wv][0][lo], dst   + e0);
    }

    int buf = 0;
    for (;;) {
        const int  tn   = t + nWaves;
        const bool more = (tn < nTiles);

        if (more) {                             // stage next tile -> other buf
            long long e0 = (long long)tn * TILE + lo;
            async_copy16(&rB[wv][buf ^ 1][lo], rdist + e0);
            async_copy16(&sB[wv][buf ^ 1][lo], src   + e0);
            async_copy16(&dB[wv][buf ^ 1][lo], dst   + e0);
            WAIT_ASYNC(3);                      // current tile's 3 ops done
        } else {
            WAIT_ASYNC(0);                      // drain
        }

        // ---- consume current buffer (ds_load_b128 x3 per lane) ------------
        float4 rq = *(const float4*)&rB[wv][buf][lo];
        int4   sq = *(const int4*)  &sB[wv][buf][lo];
        int4   dq = *(const int4*)  &dB[wv][buf][lo];

        sph_edge(rq.x, sq.x, dq.x, C, areas, rho);
        sph_edge(rq.y, sq.y, dq.y, C, areas, rho);
        sph_edge(rq.z, sq.z, dq.z, C, areas, rho);
        sph_edge(rq.w, sq.w, dq.w, C, areas, rho);

        if (!more) break;
        t = tn;
        buf ^= 1;
    }
}

__global__ void __launch_bounds__(BLOCK)
density_tail_kernel(const float* __restrict__ rdist,
                    const float* __restrict__ areas,
                    const int*   __restrict__ src,
                    const int*   __restrict__ dst,
                    const float* __restrict__ supportPtr,
                    float* __restrict__ rho,
                    int start, int E) {
    int e = start + blockIdx.x * BLOCK + threadIdx.x;
    if (e < E) {
        const float h = *supportPtr;
        const float C = (7.0f / 3.14159265358979323846f) / (h * h);
        sph_edge(rdist[e], src[e], dst[e], C, areas, rho);
    }
}

extern "C" void kernel_launch(void* const* d_in, const int* in_sizes, int n_in,
                              void* d_out, int out_size, void* d_ws, size_t ws_size,
                              hipStream_t stream) {
    (void)n_in; (void)d_ws; (void)ws_size;
    const float* rdist = (const float*)d_in[0];          // [E] f32
    const float* areas = (const float*)d_in[1];          // [N] f32
    const int*   nbr   = (const int*)  d_in[2];          // [2,E] int
    const float* sup   = (const float*)d_in[3];          // [1] f32
    float*       rho   = (float*)d_out;                  // [N] f32

    const int E = in_sizes[0];
    const int N = out_size;
    const int* src = nbr;
    const int* dst = nbr + (size_t)E;

    // rho = 0 (d_out is poisoned by the harness)
    {
        int blocks = (N + BLOCK - 1) / BLOCK;
        density_zero_kernel<<<blocks, BLOCK, 0, stream>>>(rho, N);
    }

    const int nTiles = E / TILE;
    if (nTiles > 0) {
        int blocks = (nTiles + WPB - 1) / WPB;
        if (blocks > MAX_BLOCKS) blocks = MAX_BLOCKS;    // grid-stride tiles
        density_main_kernel<<<blocks, BLOCK, 0, stream>>>(
            rdist, areas, src, dst, sup, rho, nTiles);
    }
    const int rem = E - nTiles * TILE;
    if (rem > 0) {
        int blocks = (rem + BLOCK - 1) / BLOCK;
        density_tail_kernel<<<blocks, BLOCK, 0, stream>>>(
            rdist, areas, src, dst, sup, rho, nTiles * TILE, E);
    }
}